// RNNModel_65755949302395
// MI455X (gfx1250) — compile-verified
//
#include <hip/hip_runtime.h>
#include <math.h>

// Problem dims (from reference)
#define T_ 140
#define B_ 32
#define V_ 33278
#define E_ 400
#define H_ 1150
#define HP_ 1152            // H padded to multiple of 16
#define VP_ 33280           // V padded to multiple of 16
#define TB_ (T_*B_)         // 4480 rows (multiple of 16)

typedef __attribute__((ext_vector_type(2))) float v2f;
typedef __attribute__((ext_vector_type(8))) float v8f;

// ---------------------------------------------------------------------------
// Weight padding: W [G*R, C] row-major -> Wp [G*RP, CP], per-gate row padding,
// zero-filled pads (so garbage in padded activation columns multiplies by 0).
// ---------------------------------------------------------------------------
__global__ __launch_bounds__(256) void pad_gate_weight(
    const float* __restrict__ W, float* __restrict__ Wp,
    int R, int RP, int C, int CP, long total)
{
    long idx = (long)blockIdx.x * blockDim.x + threadIdx.x;
    if (idx >= total) return;
    int  col  = (int)(idx % CP);
    long rowp = idx / CP;
    int  gate = (int)(rowp / RP);
    int  r    = (int)(rowp % RP);
    Wp[idx] = (r < R && col < C) ? W[((long)gate * R + r) * C + col] : 0.0f;
}

__global__ __launch_bounds__(256) void combine_bias(
    const float* __restrict__ bih, const float* __restrict__ bhh,
    float* __restrict__ bp, int R, int RP, int total)
{
    int idx = blockIdx.x * blockDim.x + threadIdx.x;
    if (idx >= total) return;
    int gate = idx / RP, r = idx % RP;
    bp[idx] = (r < R) ? (bih[gate * R + r] + bhh[gate * R + r]) : 0.0f;
}

// ---------------------------------------------------------------------------
// Embedding gather: x[t*B+b, e] = emb[tok[t*B+b], e]
// ---------------------------------------------------------------------------
__global__ __launch_bounds__(256) void embed_gather(
    const int* __restrict__ tok, const float* __restrict__ emb,
    float* __restrict__ x, long total)
{
    long idx = (long)blockIdx.x * blockDim.x + threadIdx.x;
    if (idx >= total) return;
    long row = idx / E_;
    int  e   = (int)(idx % E_);
    x[idx] = emb[(long)tok[row] * E_ + e];
}

// h/c init with zeroed column pads (pads stay zero through the scan).
__global__ __launch_bounds__(256) void init_hc(
    const float* __restrict__ h0, const float* __restrict__ c0,
    float* __restrict__ h, float* __restrict__ c, int H, int HP, int total)
{
    int idx = blockIdx.x * blockDim.x + threadIdx.x;
    if (idx >= total) return;
    int b = idx / HP, j = idx % HP;
    h[idx] = (j < H) ? h0[b * H + j] : 0.0f;
    c[idx] = (j < H) ? c0[b * H + j] : 0.0f;
}

// ---------------------------------------------------------------------------
// f32 WMMA GEMM:  out[M,N] = A[M,K] * W[N,K]^T + Cin
//   Cin: ldc!=0 -> matrix [M,N] (step-GEMM accumulating precomputed x-gates)
//        ldc==0 -> bias vector indexed by column (broadcast over rows)
//
// Each wave owns a 16 x (NT*16) strip: one A fragment feeds NT independent
// WMMA accumulator chains (A reuse + RAW-hazard interleaving). Block = 8
// waves arranged 2(M) x 4(N-groups) -> block tile 32 x (NT*64). Partial
// blocks are cut at wave granularity (wave-uniform return, EXEC all-1s).
//
// K-loop is a DISTANCE-2 software pipeline: the WMMAs for chunk j consume
// registers loaded two iterations earlier, while the loads for chunks j+1
// and j+2 are still in flight behind them. SIInsertWaitcnts then emits
// staged s_wait_loadcnt (<= 2*(NT+1)) instead of full drains, giving a
// ~2-chunk window of matrix-pipe time to hide the L0/L2 weight latency.
// Requires K % 4 == 0 and K >= 8 (true: 400, 1152).
//
// Fragment layouts (ISA 7.12.2):
//   A 16x4: lane l holds A[m0+(l&15)][k + (l>>4)*2 + {0,1}]   (b64 load)
//   B 4x16: lane l holds W[n0+(l&15)][k + (l>>4)*2 + {0,1}]   (b64 load)
//   C/D:    vgpr r, lane l -> D[m0 + r + (l>>4)*8][n0 + (l&15)]
// Nstore < N handles the V=33278 tail (clamped loads, guarded stores).
// ---------------------------------------------------------------------------
template<int NT>
__global__ __launch_bounds__(256) void gemm_wmma_f32(
    const float* __restrict__ A, int lda,
    const float* __restrict__ W, int ldw,
    const float* __restrict__ Cin, int ldc,
    float* __restrict__ out, int ldo,
    int M, int N, int K, int Nstore)
{
    const int lane = threadIdx.x & 31;
    const int wave = threadIdx.x >> 5;
    const int m0 = (blockIdx.y * 2 + (wave >> 2)) * 16;
    const int n0 = blockIdx.x * (4 * NT * 16) + (wave & 3) * (NT * 16);
    if (m0 >= M || n0 >= N) return;        // wave-uniform: EXEC stays all-1s

    const int tr = lane & 15;              // row-within-tile role of this lane
    const int kh = (lane >> 4) << 1;       // K phase: 0 or 2
    const int crb = m0 + ((lane >> 4) << 3);

    const float* ap = A + (size_t)(m0 + tr) * lda + kh;

    const float* wp[NT];
#pragma unroll
    for (int i = 0; i < NT; ++i) {
        int r = n0 + 16 * i + tr;
        if (r > Nstore - 1) r = Nstore - 1;
        wp[i] = W + (size_t)r * ldw + kh;
    }

    v8f acc[NT];
#pragma unroll
    for (int i = 0; i < NT; ++i) {
        int cc = n0 + 16 * i + tr;
        int ccc = (cc < Nstore) ? cc : (Nstore - 1);
#pragma unroll
        for (int r = 0; r < 8; ++r)
            acc[i][r] = ldc ? Cin[(size_t)(crb + r) * ldc + ccc] : Cin[ccc];
    }

    // ---- distance-2 software-pipelined K loop ----
    v2f avb[2];
    v2f bvb[2][NT];
#pragma unroll
    for (int s = 0; s < 2; ++s) {
        avb[s] = *(const v2f*)(ap + 4 * s);
#pragma unroll
        for (int i = 0; i < NT; ++i)
            bvb[s][i] = *(const v2f*)(wp[i] + 4 * s);
    }

#pragma unroll 2
    for (int k = 8; k < K; k += 4) {
        // prefetch chunk k (consumed two iterations from now)
        v2f av_n = *(const v2f*)(ap + k);
        v2f bv_n[NT];
#pragma unroll
        for (int i = 0; i < NT; ++i) bv_n[i] = *(const v2f*)(wp[i] + k);
        // compute chunk k-8 (its loads retired ~2 iterations ago)
#pragma unroll
        for (int i = 0; i < NT; ++i)
            acc[i] = __builtin_amdgcn_wmma_f32_16x16x4_f32(
                false, avb[0], false, bvb[0][i], (short)0, acc[i], false, false);
        // rotate buffers (pure renaming after unroll+SROA)
        avb[0] = avb[1]; avb[1] = av_n;
#pragma unroll
        for (int i = 0; i < NT; ++i) { bvb[0][i] = bvb[1][i]; bvb[1][i] = bv_n[i]; }
    }

    // drain the two remaining chunks
#pragma unroll
    for (int s = 0; s < 2; ++s)
#pragma unroll
        for (int i = 0; i < NT; ++i)
            acc[i] = __builtin_amdgcn_wmma_f32_16x16x4_f32(
                false, avb[s], false, bvb[s][i], (short)0, acc[i], false, false);

#pragma unroll
    for (int i = 0; i < NT; ++i) {
        int cc = n0 + 16 * i + tr;
        if (cc < Nstore) {
#pragma unroll
            for (int r = 0; r < 8; ++r)
                out[(size_t)(crb + r) * ldo + cc] = acc[i][r];
        }
    }
}

// ---------------------------------------------------------------------------
// LSTM pointwise update. gates: [B_, 4*HP] (i,f,g,o per-gate padded blocks).
// Writes h into the running state and into y (offset by caller for step t).
// ---------------------------------------------------------------------------
__global__ __launch_bounds__(256) void lstm_point(
    const float* __restrict__ gates, int HP, int H,
    float* __restrict__ h, float* __restrict__ c, float* __restrict__ y)
{
    int idx = blockIdx.x * blockDim.x + threadIdx.x;
    if (idx >= B_ * H) return;
    int b = idx / H, j = idx % H;
    const float* g = gates + (size_t)b * 4 * HP;
    float ig = 1.0f / (1.0f + expf(-g[j]));
    float fg = 1.0f / (1.0f + expf(-g[HP + j]));
    float gg = tanhf(g[2 * HP + j]);
    float og = 1.0f / (1.0f + expf(-g[3 * HP + j]));
    size_t o = (size_t)b * HP + j;
    float cn = fg * c[o] + ig * gg;
    float hn = og * tanhf(cn);
    c[o] = cn; h[o] = hn; y[o] = hn;
}

// ---------------------------------------------------------------------------
static inline int cdiv(long a, int b) { return (int)((a + b - 1) / b); }

extern "C" void kernel_launch(void* const* d_in, const int* in_sizes, int n_in,
                              void* d_out, int out_size, void* d_ws, size_t ws_size,
                              hipStream_t stream)
{
    const int*   tok  = (const int*)  d_in[0];
    const float* h0_0 = (const float*)d_in[1];
    const float* c0_0 = (const float*)d_in[2];
    const float* h0_1 = (const float*)d_in[3];
    const float* c0_1 = (const float*)d_in[4];
    const float* h0_2 = (const float*)d_in[5];
    const float* c0_2 = (const float*)d_in[6];
    const float* emb  = (const float*)d_in[7];
    const float* Wih0 = (const float*)d_in[8];
    const float* Whh0 = (const float*)d_in[9];
    const float* bih0 = (const float*)d_in[10];
    const float* bhh0 = (const float*)d_in[11];
    const float* Wih1 = (const float*)d_in[12];
    const float* Whh1 = (const float*)d_in[13];
    const float* bih1 = (const float*)d_in[14];
    const float* bhh1 = (const float*)d_in[15];
    const float* Wih2 = (const float*)d_in[16];
    const float* Whh2 = (const float*)d_in[17];   // [1600,400]: already 16-aligned
    const float* bih2 = (const float*)d_in[18];
    const float* bhh2 = (const float*)d_in[19];
    const float* decb = (const float*)d_in[20];
    float* out = (float*)d_out;

    // ---- workspace carve-up (floats); total ~190 MB ----
    float* ws = (float*)d_ws;
    size_t off = 0;
    auto alloc = [&](size_t n) { float* p = ws + off; off += n; return p; };
    float* Wih0p = alloc((size_t)4 * HP_ * E_);     // [4608,400]
    float* Whh0p = alloc((size_t)4 * HP_ * HP_);    // [4608,1152]
    float* Wih1p = alloc((size_t)4 * HP_ * HP_);
    float* Whh1p = alloc((size_t)4 * HP_ * HP_);
    float* Wih2p = alloc((size_t)4 * E_ * HP_);     // [1600,1152]
    float* b0p   = alloc(4 * HP_);
    float* b1p   = alloc(4 * HP_);
    float* b2p   = alloc(4 * E_);
    float* x0    = alloc((size_t)TB_ * E_);         // embed output; later y2
    float* ybuf  = alloc((size_t)TB_ * HP_);        // y0, then y1 (sequential reuse)
    float* XG    = alloc((size_t)TB_ * 4 * HP_);    // precomputed input gates
    float* gates = alloc((size_t)B_ * 4 * HP_);
    float* hbuf  = alloc((size_t)B_ * HP_);
    float* cbuf  = alloc((size_t)B_ * HP_);
    (void)off; (void)ws_size; (void)in_sizes; (void)n_in; (void)out_size;

    const int TPB = 256;

    // ---- weight/bias padding ----
    { long n = (long)4 * HP_ * E_;
      pad_gate_weight<<<cdiv(n, TPB), TPB, 0, stream>>>(Wih0, Wih0p, H_, HP_, E_, E_, n); }
    { long n = (long)4 * HP_ * HP_;
      pad_gate_weight<<<cdiv(n, TPB), TPB, 0, stream>>>(Whh0, Whh0p, H_, HP_, H_, HP_, n);
      pad_gate_weight<<<cdiv(n, TPB), TPB, 0, stream>>>(Wih1, Wih1p, H_, HP_, H_, HP_, n);
      pad_gate_weight<<<cdiv(n, TPB), TPB, 0, stream>>>(Whh1, Whh1p, H_, HP_, H_, HP_, n); }
    { long n = (long)4 * E_ * HP_;
      pad_gate_weight<<<cdiv(n, TPB), TPB, 0, stream>>>(Wih2, Wih2p, E_, E_, H_, HP_, n); }
    combine_bias<<<cdiv(4 * HP_, TPB), TPB, 0, stream>>>(bih0, bhh0, b0p, H_, HP_, 4 * HP_);
    combine_bias<<<cdiv(4 * HP_, TPB), TPB, 0, stream>>>(bih1, bhh1, b1p, H_, HP_, 4 * HP_);
    combine_bias<<<cdiv(4 * E_,  TPB), TPB, 0, stream>>>(bih2, bhh2, b2p, E_, E_, 4 * E_);

    // ---- embedding ----
    { long n = (long)TB_ * E_;
      embed_gather<<<cdiv(n, TPB), TPB, 0, stream>>>(tok, emb, x0, n); }

    // ==================== Layer 0 (E=400 -> H=1150) ====================
    // XG = x0 * Wih0^T + (bih0+bhh0)   [4480, 4608]
    gemm_wmma_f32<4><<<dim3(cdiv(4 * HP_, 256), TB_ / 32), TPB, 0, stream>>>(
        x0, E_, Wih0p, E_, b0p, 0, XG, 4 * HP_, TB_, 4 * HP_, E_, 4 * HP_);
    init_hc<<<cdiv(B_ * HP_, TPB), TPB, 0, stream>>>(h0_0, c0_0, hbuf, cbuf, H_, HP_, B_ * HP_);
    for (int t = 0; t < T_; ++t) {
        gemm_wmma_f32<2><<<dim3(cdiv(4 * HP_, 128), 1), TPB, 0, stream>>>(
            hbuf, HP_, Whh0p, HP_, XG + (size_t)t * B_ * 4 * HP_, 4 * HP_,
            gates, 4 * HP_, B_, 4 * HP_, HP_, 4 * HP_);
        lstm_point<<<cdiv(B_ * H_, TPB), TPB, 0, stream>>>(
            gates, HP_, H_, hbuf, cbuf, ybuf + (size_t)t * B_ * HP_);
    }

    // ==================== Layer 1 (H -> H) ====================
    gemm_wmma_f32<4><<<dim3(cdiv(4 * HP_, 256), TB_ / 32), TPB, 0, stream>>>(
        ybuf, HP_, Wih1p, HP_, b1p, 0, XG, 4 * HP_, TB_, 4 * HP_, HP_, 4 * HP_);
    init_hc<<<cdiv(B_ * HP_, TPB), TPB, 0, stream>>>(h0_1, c0_1, hbuf, cbuf, H_, HP_, B_ * HP_);
    for (int t = 0; t < T_; ++t) {
        gemm_wmma_f32<2><<<dim3(cdiv(4 * HP_, 128), 1), TPB, 0, stream>>>(
            hbuf, HP_, Whh1p, HP_, XG + (size_t)t * B_ * 4 * HP_, 4 * HP_,
            gates, 4 * HP_, B_, 4 * HP_, HP_, 4 * HP_);
        lstm_point<<<cdiv(B_ * H_, TPB), TPB, 0, stream>>>(
            gates, HP_, H_, hbuf, cbuf, ybuf + (size_t)t * B_ * HP_);  // y1 over y0
    }

    // ==================== Layer 2 (H -> E=400) ====================
    gemm_wmma_f32<4><<<dim3(cdiv(4 * E_, 256), TB_ / 32), TPB, 0, stream>>>(
        ybuf, HP_, Wih2p, HP_, b2p, 0, XG, 4 * E_, TB_, 4 * E_, HP_, 4 * E_);
    init_hc<<<cdiv(B_ * E_, TPB), TPB, 0, stream>>>(h0_2, c0_2, hbuf, cbuf, E_, E_, B_ * E_);
    for (int t = 0; t < T_; ++t) {
        gemm_wmma_f32<2><<<dim3(cdiv(4 * E_, 128), 1), TPB, 0, stream>>>(
            hbuf, E_, Whh2, E_, XG + (size_t)t * B_ * 4 * E_, 4 * E_,
            gates, 4 * E_, B_, 4 * E_, E_, 4 * E_);
        lstm_point<<<cdiv(B_ * E_, TPB), TPB, 0, stream>>>(
            gates, E_, E_, hbuf, cbuf, x0 + (size_t)t * B_ * E_);      // y2 into x0 slot
    }

    // ==================== Tied decoder: out = y2 * emb^T + dec_b ====================
    gemm_wmma_f32<4><<<dim3(VP_ / 256, TB_ / 32), TPB, 0, stream>>>(
        x0, E_, emb, E_, decb, 0, out, V_, TB_, VP_, E_, V_);
}